// Test_20675972563815
// MI455X (gfx1250) — compile-verified
//
#include <hip/hip_runtime.h>

// Reference: x,y are [T, NTR, CH] f32, T=5000, NTR=1024, CH=2.
// Flat batch B=2048; element (t,b) at t*B+b (contiguous in b -> coalesced).
#define T_LEN 5000
#define BATCH 2048
#define CS    200          // chunk size along T
#define NC    25           // chunks; CS*NC == T_LEN
#define B2    (2*BATCH)    // (trace,channel,sign) problems
#define STRIP 8            // rows per async-staged LDS strip in pass 3
#define NSTRIP (CS/STRIP)  // 25 strips per chunk

// Workspace (floats):
//  [0, 4*NC*BATCH)       chunk partial sums -> (after pass2) EXCLUSIVE prefixes
//                        q = 0:x_pos 1:x_neg 2:y_pos 3:y_neg, ((q*NC+c)*BATCH+b)
//  [TOT_OFF, +4*BATCH)   totals (normalizers)
//  [BSUM_OFF, +NBLK3)    per-block partial losses
#define PART_SZ  (4*NC*BATCH)
#define TOT_OFF  PART_SZ
#define BSUM_OFF (TOT_OFF + 4*BATCH)
#define NBLK3    ((NC*B2)/256)     // 400

typedef __attribute__((ext_vector_type(2))) float v2f;
typedef __attribute__((ext_vector_type(8))) float v8f;

// Exact payload type from the builtin's prototype (from the round-2 diagnostic):
// 'int __attribute__((vector_size(16)))' with AS1 (global) / AS3 (LDS) pointers.
typedef int v4i_b128 __attribute__((vector_size(4 * sizeof(int))));
typedef __attribute__((address_space(1))) v4i_b128* as1_v4i_ptr;
typedef __attribute__((address_space(3))) v4i_b128* as3_v4i_ptr;

#if defined(__gfx1250__) && __has_builtin(__builtin_amdgcn_global_load_async_to_lds_b128)
#define USE_ASYNC_LDS 1
#else
#define USE_ASYNC_LDS 0
#endif

__device__ __forceinline__ void wait_async_le2() {
#if __has_builtin(__builtin_amdgcn_s_wait_asynccnt)
    __builtin_amdgcn_s_wait_asynccnt(2);
#else
    asm volatile("s_wait_asynccnt 0x2" ::: "memory");
#endif
}
__device__ __forceinline__ void wait_async_le0() {
#if __has_builtin(__builtin_amdgcn_s_wait_asynccnt)
    __builtin_amdgcn_s_wait_asynccnt(0);
#else
    asm volatile("s_wait_asynccnt 0x0" ::: "memory");
#endif
}

// ---------------------------------------------------------------------------
// Pass 1: per-(chunk,batch) partial sums of max(+-x,0), max(+-y,0).
// float4 lanes (global_load_b128): 4x fewer VMEM instructions so the pass is
// HBM-BW-bound (~82 MB @ 23.3 TB/s ~ 3.5us), not issue-bound; warms the
// 192 MB L2 (x+y = 82 MB fit entirely) for pass 3.
// ---------------------------------------------------------------------------
__device__ __forceinline__ void acc4(float4 v, float4& p, float4& n) {
    p.x += fmaxf(v.x, 0.f);  n.x += fmaxf(-v.x, 0.f);
    p.y += fmaxf(v.y, 0.f);  n.y += fmaxf(-v.y, 0.f);
    p.z += fmaxf(v.z, 0.f);  n.z += fmaxf(-v.z, 0.f);
    p.w += fmaxf(v.w, 0.f);  n.w += fmaxf(-v.w, 0.f);
}

__global__ void ot_pass1(const float* __restrict__ x, const float* __restrict__ y,
                         float* __restrict__ ws) {
    const int BQ = BATCH / 4;                         // 512 float4 per row
    int u  = blockIdx.x * blockDim.x + threadIdx.x;   // < NC*BQ
    int bq = u & (BQ - 1);
    int c  = u >> 9;
    const float4* xp = (const float4*)x + (size_t)c * CS * BQ + bq;
    const float4* yp = (const float4*)y + (size_t)c * CS * BQ + bq;
    float4 a0 = {0,0,0,0}, a1 = {0,0,0,0}, a2 = {0,0,0,0}, a3 = {0,0,0,0};
    for (int tt = 0; tt < CS; ++tt) {
        float4 xv = xp[(size_t)tt * BQ];
        float4 yv = yp[(size_t)tt * BQ];
        acc4(xv, a0, a1);
        acc4(yv, a2, a3);
    }
    float4* w4 = (float4*)ws;
    w4[((0 * NC + c) * BATCH >> 2) + bq] = a0;
    w4[((1 * NC + c) * BATCH >> 2) + bq] = a1;
    w4[((2 * NC + c) * BATCH >> 2) + bq] = a2;
    w4[((3 * NC + c) * BATCH >> 2) + bq] = a3;
}

// ---------------------------------------------------------------------------
// Pass 2: exclusive prefix over the 25 chunk sums (in place) + totals.
// ---------------------------------------------------------------------------
__global__ void ot_pass2(float* __restrict__ ws) {
    int u = blockIdx.x * blockDim.x + threadIdx.x;    // < 4*BATCH
    int b = u & (BATCH - 1);
    int q = u >> 11;
    float run = 0.f;
    for (int c = 0; c < NC; ++c) {
        int idx = (q * NC + c) * BATCH + b;
        float p = ws[idx];
        ws[idx] = run;
        run += p;
    }
    ws[TOT_OFF + q * BATCH + b] = run;
}

// ---------------------------------------------------------------------------
// Pass 3 merge step: idx = searchsorted(G, F, 'left')-1 clipped to [0,T-2],
// maintained by a monotone two-pointer walk (the serial y chain is a true
// dependency; it hits L2 since x+y are resident there after pass 1).
// ---------------------------------------------------------------------------
__device__ __forceinline__ void merge_step(int t, float xv, int sign,
                                           float invsf, float invsg,
                                           const float* __restrict__ y, int b,
                                           int& j, float& G_lo, float& G_hi,
                                           float& Fn, float& loss) {
    float f = fmaxf(sign ? -xv : xv, 0.f) * invsf;
    Fn += f;
    while (j < T_LEN - 2 && G_hi < Fn) {
        ++j;
        float yv = y[(size_t)(j + 1) * BATCH + b];
        G_lo = G_hi;
        G_hi += fmaxf(sign ? -yv : yv, 0.f) * invsg;
    }
    float frac = (Fn - G_lo) / (G_hi - G_lo + 1e-10f);
    float d    = (float)t - ((float)j + frac);
    loss = fmaf(d * d, f, loss);
}

// ---------------------------------------------------------------------------
// Pass 3: thread = (b, sign, chunk).  F resumes from the x prefix; G pointer
// resumes from the largest G-chunk-prefix strictly below F_start (valid lower
// bound for the walk).  The block's 200x256 x-tile is streamed through LDS
// with double-buffered GLOBAL_LOAD_ASYNC_TO_LDS_B128 strips (ASYNCcnt paced).
// Block reduction = WMMA ones-matrix column sums + ds_bpermute fold.
// ---------------------------------------------------------------------------
__global__ void ot_pass3(const float* __restrict__ x, const float* __restrict__ y,
                         float* __restrict__ ws) {
#if USE_ASYNC_LDS
    __shared__ __align__(16) float xtile[2][STRIP * 256];   // 2 x 8 KB
#endif
    __shared__ float sdata[256];

    const int tid   = threadIdx.x;
    const int bbase = (blockIdx.x & 7) * 256;
    const int b     = bbase + tid;
    const int sign  = (blockIdx.x >> 3) & 1;
    const int c     = blockIdx.x >> 4;                // chunk 0..24

    float sf    = ws[TOT_OFF + sign * BATCH + b];
    float sg    = ws[TOT_OFF + (2 + sign) * BATCH + b];
    float invsf = 1.0f / sf;
    float invsg = 1.0f / sg;

    const float* preff = ws + (size_t)(sign)     * NC * BATCH + b;
    const float* prefg = ws + (size_t)(2 + sign) * NC * BATCH + b;

    float Fstart = preff[(size_t)c * BATCH] * invsf;

    int cg = 0;
    for (int k = NC - 1; k >= 1; --k) {
        if (prefg[(size_t)k * BATCH] * invsg < Fstart) { cg = k; break; }
    }

    int j; float G_lo, G_hi;
    if (cg == 0) {
        j = 0;
        float y0 = y[b];
        float y1 = y[BATCH + b];
        G_lo = fmaxf(sign ? -y0 : y0, 0.f) * invsg;
        G_hi = G_lo + fmaxf(sign ? -y1 : y1, 0.f) * invsg;
    } else {
        j    = cg * CS - 1;                           // G[j] = prefg[cg] < Fstart
        G_lo = prefg[(size_t)cg * BATCH] * invsg;
        float yv = y[(size_t)(j + 1) * BATCH + b];
        G_hi = G_lo + fmaxf(sign ? -yv : yv, 0.f) * invsg;
    }

    const int t0 = c * CS;
    float Fn = Fstart, loss = 0.f;

#if USE_ASYNC_LDS
    auto issue_strip = [&](int s, int bufi) {
        // strip = 8 rows x 256 cols; each thread async-copies 2 x 16B
        int row  = tid >> 6;                          // 0..3
        int col4 = (tid & 63) << 2;                   // 0,4,...,252
        const float* g0 = x + (size_t)(t0 + s * STRIP + row) * BATCH + bbase + col4;
        const float* g1 = g0 + (size_t)4 * BATCH;
        __builtin_amdgcn_global_load_async_to_lds_b128(
            (as1_v4i_ptr)g0,
            (as3_v4i_ptr)&xtile[bufi][row * 256 + col4], 0, 0);
        __builtin_amdgcn_global_load_async_to_lds_b128(
            (as1_v4i_ptr)g1,
            (as3_v4i_ptr)&xtile[bufi][(row + 4) * 256 + col4], 0, 0);
    };
    issue_strip(0, 0);
    issue_strip(1, 1);
    for (int s = 0; s < NSTRIP; ++s) {
        int buf = s & 1;
        if (s + 1 < NSTRIP) wait_async_le2();         // oldest 2 (strip s) done
        else                wait_async_le0();
        __syncthreads();                              // publish strip s to all waves
        for (int r = 0; r < STRIP; ++r) {
            int t = t0 + s * STRIP + r;
            merge_step(t, xtile[buf][r * 256 + tid], sign, invsf, invsg,
                       y, b, j, G_lo, G_hi, Fn, loss);
        }
        __syncthreads();                              // everyone done reading buf
        if (s + 2 < NSTRIP) issue_strip(s + 2, buf);  // refill freed buffer
    }
#else
    for (int tt = 0; tt < CS; ++tt) {
        int t = t0 + tt;
        float xv = x[(size_t)t * BATCH + b];
        if (tt + 24 < CS)
            __builtin_prefetch(x + (size_t)(t + 24) * BATCH + b, 0, 1);
        merge_step(t, xv, sign, invsf, invsg, y, b, j, G_lo, G_hi, Fn, loss);
    }
#endif

    sdata[tid] = loss;
    __syncthreads();

#if __has_builtin(__builtin_amdgcn_wmma_f32_16x16x4_f32)
    // D = ones(16x4) x B(4x16) + C: 4 accumulating WMMAs fold 256 partials
    // into 16 column sums; 4 shfl_xor steps finish.  Wave-uniform branch.
    if (tid < 32) {
        int lane = tid;
        v8f acc = {0.f, 0.f, 0.f, 0.f, 0.f, 0.f, 0.f, 0.f};
        v2f av; av[0] = 1.0f; av[1] = 1.0f;
        for (int q = 0; q < 4; ++q) {
            int base = 64 * q + (lane & 15) + ((lane >> 4) << 5);
            v2f bv; bv[0] = sdata[base]; bv[1] = sdata[base + 16];
            acc = __builtin_amdgcn_wmma_f32_16x16x4_f32(
                false, av, false, bv, (short)0, acc, false, false);
        }
        float col = acc[0];
        col += __shfl_xor(col, 8, 32);
        col += __shfl_xor(col, 4, 32);
        col += __shfl_xor(col, 2, 32);
        col += __shfl_xor(col, 1, 32);
        if (lane == 0) ws[BSUM_OFF + blockIdx.x] = col;
    }
#else
    for (int s = 128; s > 0; s >>= 1) {
        if (tid < s) sdata[tid] += sdata[tid + s];
        __syncthreads();
    }
    if (tid == 0) ws[BSUM_OFF + blockIdx.x] = sdata[0];
#endif
}

// ---------------------------------------------------------------------------
// Pass 4: fold 400 block sums into the scalar (fixed tree, deterministic).
// ---------------------------------------------------------------------------
__global__ void ot_pass4(const float* __restrict__ bsum, float* __restrict__ out) {
    __shared__ float sdata[256];
    float v = 0.f;
    for (int i = threadIdx.x; i < NBLK3; i += 256) v += bsum[i];
    sdata[threadIdx.x] = v;
    __syncthreads();
    for (int s = 128; s > 0; s >>= 1) {
        if (threadIdx.x < s) sdata[threadIdx.x] += sdata[threadIdx.x + s];
        __syncthreads();
    }
    if (threadIdx.x == 0) out[0] = sdata[0];
}

extern "C" void kernel_launch(void* const* d_in, const int* in_sizes, int n_in,
                              void* d_out, int out_size, void* d_ws, size_t ws_size,
                              hipStream_t stream) {
    const float* x = (const float*)d_in[0];
    const float* y = (const float*)d_in[1];
    float* ws  = (float*)d_ws;     // ~853 KB scratch
    float* out = (float*)d_out;

    ot_pass1<<<(NC * BATCH / 4) / 256, 256, 0, stream>>>(x, y, ws);  // 50 blocks
    ot_pass2<<<(4 * BATCH) / 256, 256, 0, stream>>>(ws);             // 32 blocks
    ot_pass3<<<NBLK3, 256, 0, stream>>>(x, y, ws);                   // 400 blocks
    ot_pass4<<<1, 256, 0, stream>>>(ws + BSUM_OFF, out);
}